// SAGEConvAggLoc_18580028522748
// MI455X (gfx1250) — compile-verified
//
#include <hip/hip_runtime.h>
#include <hip/hip_bf16.h>

// GraphSAGE mean aggregation for MI455X (gfx1250, wave32).
// CSR-by-destination build (counts -> single-WG scan -> bucket scatter), then
// atomic-free aggregation with one wave per node. The per-block edge-index
// range is staged into LDS with the Tensor Data Mover (tensor_load_to_lds,
// TENSORcnt), so the hot loop reads indices from LDS and only the coalesced
// 256B feature-row gathers (L2-resident: 25.6 MB table vs 192 MB L2) go to
// the vector memory path. Edge streams use NT temporal hints.

#define GS_N_NODES 100000
#define GS_N_EDGES 1000000
#define GS_D_FEAT  64
#define GS_TDM_CAP 2048   // staged edge indices per block (8 KB LDS)

// --- TDM builtin arity differs between toolchains (see CDNA5_HIP.md). ---
#if __has_include(<hip/amd_detail/amd_gfx1250_TDM.h>)
#define GS_TDM_6ARG 1     // therock/clang-23: 6-arg builtin
#else
#define GS_TDM_6ARG 0     // ROCm 7.2/clang-22: 5-arg builtin
#endif

typedef __attribute__((ext_vector_type(4))) unsigned int gs_v4u;
typedef __attribute__((ext_vector_type(8))) int          gs_v8i;
typedef __attribute__((ext_vector_type(4))) int          gs_v4i;

// One-shot TDM copy: cnt_dwords contiguous dwords from global -> LDS.
// Descriptor packed per CDNA5 ISA D# layout (groups 0/1; groups 2/3 unused).
__device__ __forceinline__ void gs_tdm_copy_to_lds(unsigned lds_off,
                                                   unsigned long long gaddr,
                                                   unsigned cnt_dwords) {
    const unsigned ga_lo = (unsigned)(gaddr & 0xFFFFFFFFull);
    const unsigned ga_hi = (unsigned)(gaddr >> 32);

    gs_v4u g0;
    g0.x = 1u;                                   // count=1 (valid), no gather
    g0.y = lds_off;                              // lds_addr (bytes)
    g0.z = ga_lo;                                // global_addr[31:0]
    g0.w = (ga_hi & 0x01FFFFFFu) | (2u << 30);   // global_addr[56:32] | type=2

    gs_v8i g1;
    g1[0] = (int)(2u << 16);                               // data_size=4B
    g1[1] = (int)((cnt_dwords & 0xFFFFu) << 16);           // tensor_dim0[15:0]
    g1[2] = (int)(((cnt_dwords >> 16) & 0xFFFFu) | (1u << 16)); // dim0 hi | tensor_dim1=1
    g1[3] = (int)((cnt_dwords & 0xFFFFu) << 16);           // tile_dim0 = cnt
    g1[4] = 1;                                             // tile_dim1=1, tile_dim2=0
    g1[5] = (int)cnt_dwords;                               // tensor_dim0_stride lo
    g1[6] = 0;
    g1[7] = 0;

    gs_v4i gz = {0, 0, 0, 0};
#if GS_TDM_6ARG
    gs_v8i gz8 = {0, 0, 0, 0, 0, 0, 0, 0};
    __builtin_amdgcn_tensor_load_to_lds(g0, g1, gz, gz, gz8, 0);
#else
    __builtin_amdgcn_tensor_load_to_lds(g0, g1, gz, gz, 0);
#endif
}

__global__ void gs_zero_u32(unsigned* __restrict__ p, int n) {
    int i = blockIdx.x * blockDim.x + threadIdx.x;
    if (i < n) p[i] = 0u;
}

__global__ void gs_zero_f32(float* __restrict__ p, int n) {
    int i = blockIdx.x * blockDim.x + threadIdx.x;
    if (i < n) p[i] = 0.0f;
}

// Pass 1: in-degree counts. dst is streamed once -> non-temporal loads.
__global__ void gs_count_deg(const long long* __restrict__ dst,
                             unsigned* __restrict__ counts, int e) {
    int i = blockIdx.x * blockDim.x + threadIdx.x;
    if (i < e) {
        long long v = __builtin_nontemporal_load(dst + i);
        atomicAdd(&counts[(int)v], 1u);
    }
}

// Pass 2: exclusive scan of 100K counts in a single 1024-thread workgroup.
__global__ __launch_bounds__(1024) void gs_scan_offsets(
    const unsigned* __restrict__ counts, unsigned* __restrict__ offsets,
    unsigned* __restrict__ cursor, int n) {
    __shared__ unsigned sums[1024];
    const int T = 1024;
    const int t = threadIdx.x;
    const int chunk = (n + T - 1) / T;
    const int beg = t * chunk;
    const int end = (beg + chunk < n) ? (beg + chunk) : n;

    unsigned s = 0u;
    for (int i = beg; i < end; ++i) s += counts[i];
    sums[t] = s;
    __syncthreads();

    for (int off = 1; off < T; off <<= 1) {
        unsigned v = (t >= off) ? sums[t - off] : 0u;
        __syncthreads();
        sums[t] += v;
        __syncthreads();
    }

    unsigned run = (t == 0) ? 0u : sums[t - 1];
    for (int i = beg; i < end; ++i) {
        offsets[i] = run;
        cursor[i]  = run;
        run += counts[i];
    }
    if (t == 0) offsets[n] = sums[T - 1];
}

// Pass 3: scatter source ids into per-destination buckets.
__global__ void gs_fill_edges(const long long* __restrict__ src,
                              const long long* __restrict__ dst,
                              unsigned* __restrict__ cursor,
                              unsigned* __restrict__ edge_src, int e) {
    int i = blockIdx.x * blockDim.x + threadIdx.x;
    if (i < e) {
        long long u = __builtin_nontemporal_load(src + i);
        long long v = __builtin_nontemporal_load(dst + i);
        unsigned pos = atomicAdd(&cursor[(int)v], 1u);
        edge_src[pos] = (unsigned)u;
    }
}

// Pass 4: one wave32 per node; each lane owns a float2 slice of the row.
// The block's contiguous edge-index range is DMA'd into LDS by the TDM,
// so index reads are ds_loads and global traffic is only the coalesced
// 256B row gathers (+ prefetch of the next row).
__global__ __launch_bounds__(256) void gs_aggregate(
    const float* __restrict__ feat, const unsigned* __restrict__ offsets,
    const unsigned* __restrict__ edge_src, float* __restrict__ out, int n) {
    __shared__ unsigned shidx[GS_TDM_CAP];

    const int lane   = threadIdx.x & 31;
    const int wv     = threadIdx.x >> 5;
    const int vfirst = blockIdx.x * 8;
    const int vlastn = (vfirst + 8 < n) ? (vfirst + 8) : n;
    const int v      = vfirst + wv;

    unsigned blockBeg = offsets[vfirst];
    unsigned blockEnd = offsets[vlastn];
    blockBeg = (unsigned)__builtin_amdgcn_readfirstlane((int)blockBeg);
    blockEnd = (unsigned)__builtin_amdgcn_readfirstlane((int)blockEnd);
    const unsigned blockCnt = blockEnd - blockBeg;
    const bool useLds = (blockCnt > 0u) && (blockCnt <= (unsigned)GS_TDM_CAP);

    if (useLds && wv == 0) {
        const unsigned lds_off =
            (unsigned)(unsigned long long)(uintptr_t)(void*)shidx;
        gs_tdm_copy_to_lds(lds_off,
                           (unsigned long long)(uintptr_t)(edge_src + blockBeg),
                           blockCnt);
        __builtin_amdgcn_s_wait_tensorcnt(0);  // DMA complete before barrier
    }
    __syncthreads();

    if (v < n) {
        const unsigned beg = offsets[v];
        const unsigned end = offsets[v + 1];
        float2 acc; acc.x = 0.0f; acc.y = 0.0f;

        if (useLds) {
            if (beg < end) {
                unsigned u = shidx[beg - blockBeg];
                for (unsigned j = beg; j < end; ++j) {
                    const float2* row =
                        (const float2*)(feat + (size_t)u * GS_D_FEAT);
                    unsigned unext =
                        (j + 1u < end) ? shidx[j + 1u - blockBeg] : u;
                    __builtin_prefetch(feat + (size_t)unext * GS_D_FEAT, 0, 3);
                    float2 x = row[lane];
                    acc.x += x.x;
                    acc.y += x.y;
                    u = unext;
                }
            }
        } else if (beg < end) {
            unsigned u = __builtin_nontemporal_load(edge_src + beg);
            for (unsigned j = beg; j < end; ++j) {
                const float2* row = (const float2*)(feat + (size_t)u * GS_D_FEAT);
                unsigned unext = (j + 1u < end)
                                     ? __builtin_nontemporal_load(edge_src + j + 1u)
                                     : u;
                __builtin_prefetch(feat + (size_t)unext * GS_D_FEAT, 0, 3);
                float2 x = row[lane];
                acc.x += x.x;
                acc.y += x.y;
                u = unext;
            }
        }

        const float deg   = (float)(end - beg);
        const float scale = (end > beg) ? (1.0f / deg) : 0.0f;
        float2 o; o.x = acc.x * scale; o.y = acc.y * scale;
        ((float2*)(out + (size_t)v * GS_D_FEAT))[lane] = o;
    }
}

// ---- Fallback (workspace too small for CSR): direct atomic scatter. ----
__global__ void gs_scatter_atomic(const float* __restrict__ feat,
                                  const long long* __restrict__ src,
                                  const long long* __restrict__ dst,
                                  float* __restrict__ out,
                                  float* __restrict__ deg, int e) {
    int tid  = blockIdx.x * blockDim.x + threadIdx.x;
    int edge = tid >> 4;
    int l    = tid & 15;
    if (edge < e) {
        long long u = __builtin_nontemporal_load(src + edge);
        long long v = __builtin_nontemporal_load(dst + edge);
        const float4 x = ((const float4*)(feat + (size_t)u * GS_D_FEAT))[l];
        float* o = out + (size_t)v * GS_D_FEAT + l * 4;
        atomicAdd(o + 0, x.x);
        atomicAdd(o + 1, x.y);
        atomicAdd(o + 2, x.z);
        atomicAdd(o + 3, x.w);
        if (l == 0) atomicAdd(&deg[(int)v], 1.0f);
    }
}

__global__ void gs_normalize(float* __restrict__ out,
                             const float* __restrict__ deg, int total) {
    int i = blockIdx.x * blockDim.x + threadIdx.x;
    if (i < total) {
        float d = deg[i >> 6];
        float s = out[i];
        out[i] = (d > 0.0f) ? (s / d) : 0.0f;
    }
}

extern "C" void kernel_launch(void* const* d_in, const int* in_sizes, int n_in,
                              void* d_out, int out_size, void* d_ws, size_t ws_size,
                              hipStream_t stream) {
    const float*     feat = (const float*)d_in[0];
    const long long* src  = (const long long*)d_in[1];
    const long long* dst  = (const long long*)d_in[2];
    float*           out  = (float*)d_out;

    const int n = GS_N_NODES;
    const int e = GS_N_EDGES;

    const size_t csr_bytes = (size_t)(3 * n + 1 + e) * sizeof(unsigned);

    if (ws_size >= csr_bytes) {
        unsigned* counts   = (unsigned*)d_ws;
        unsigned* offsets  = counts + n;         // n+1 entries
        unsigned* cursor   = offsets + (n + 1);  // n entries
        unsigned* edge_src = cursor + n;         // e entries

        gs_zero_u32<<<(n + 255) / 256, 256, 0, stream>>>(counts, n);
        gs_count_deg<<<(e + 255) / 256, 256, 0, stream>>>(dst, counts, e);
        gs_scan_offsets<<<1, 1024, 0, stream>>>(counts, offsets, cursor, n);
        gs_fill_edges<<<(e + 255) / 256, 256, 0, stream>>>(src, dst, cursor,
                                                           edge_src, e);
        gs_aggregate<<<(n + 7) / 8, 256, 0, stream>>>(feat, offsets, edge_src,
                                                      out, n);
    } else {
        float* deg = (float*)d_ws;
        const int total = n * GS_D_FEAT;
        gs_zero_f32<<<(total + 255) / 256, 256, 0, stream>>>(out, total);
        gs_zero_f32<<<(n + 255) / 256, 256, 0, stream>>>(deg, n);
        const int threads = e * 16;
        gs_scatter_atomic<<<(threads + 255) / 256, 256, 0, stream>>>(
            feat, src, dst, out, deg, e);
        gs_normalize<<<(total + 255) / 256, 256, 0, stream>>>(out, deg, total);
    }
}